// Render_45775761440943
// MI455X (gfx1250) — compile-verified
//
#include <hip/hip_runtime.h>

typedef float v2f __attribute__((ext_vector_type(2)));
typedef float v8f __attribute__((ext_vector_type(8)));

__device__ __forceinline__ float fast_rcp(float x) {
    float r = __builtin_amdgcn_rcpf(x);
    return fmaf(fmaf(-x, r, 1.0f), r, r);        // 1 Newton step
}
__device__ __forceinline__ float fast_rsqrt(float x) {
    float r = __builtin_amdgcn_rsqf(x);
    return r * fmaf(-0.5f * x, r * r, 1.5f);     // 1 Newton step
}
__device__ __forceinline__ float lane_shuffle(int srcLane, float v) {
    return __int_as_float(
        __builtin_amdgcn_ds_bpermute(srcLane << 2, __float_as_int(v)));
}

// ws layout per batch (16 floats):
// [0..3] = tf row0 (R00 R01 R02 t0), [4..7] = row1, [8..11] = row2,
// [12] fx, [13] fy, [14] limx = 1.3*0.5*w/fx, [15] limy
__global__ void cam_setup_kernel(const int* __restrict__ indexes,
                                 const float* __restrict__ tf,
                                 const float* __restrict__ focal,
                                 const float* __restrict__ widths,
                                 float* __restrict__ cams, int batch) {
    int b = threadIdx.x;
    if (b < batch) {
        int idx = indexes[b];
        const float* t = tf + (size_t)idx * 16;
        float* o = cams + b * 16;
#pragma unroll
        for (int i = 0; i < 12; ++i) o[i] = t[i];   // [R | t] rows 0..2
        float fx = focal[idx * 2 + 0];
        float fy = focal[idx * 2 + 1];
        float w  = widths[idx];
        o[12] = fx;
        o[13] = fy;
        o[14] = 1.3f * 0.5f * w / fx;
        o[15] = 1.3f * 0.5f * w / fy;
    }
}

// One wave handles 32 gaussians x 4 cameras, in two 16-gaussian phases.
// A (16x4 f32): rows 4c+r = camera c's [R|t] row r; rows 4c+3 = 0.
//   WMMA A layout: lane L holds A[M=L%16][K=2*(L/16)+v] in VGPR v.
// B (4x16): column N = [x,y,z,1] of gaussian tile+N.
//   Layout: lane L holds B[K=2*(L/16)+v][N=L%16].
// D (16x16): lane L, VGPR r = D[M=r+8*(L/16)][N=L%16]
//   => lanes 0-15: means cam0 in d0..d2, cam1 in d4..d6
//      lanes 16-31: means cam2 in d0..d2, cam3 in d4..d6
// Each lane owns gaussian base+lane for cov3d; phases pull the needed
// cov3d values across lanes with ds_bpermute.
__global__ __launch_bounds__(256) void
splat_cov2d_kernel(const float* __restrict__ xyz,
                   const float* __restrict__ scales,
                   const float* __restrict__ quats,
                   const float* __restrict__ cams,
                   float* __restrict__ out, int n) {
    const int lane   = threadIdx.x & 31;
    const int waveId = (blockIdx.x * blockDim.x + threadIdx.x) >> 5;
    int base = waveId * 32;
    if (base > n - 32) base = n - 32;   // wave-uniform clamp; tail waves overlap

    const int m  = lane & 15;
    const int kb = (lane >> 4) << 1;    // K base for A/B: 0 or 2
    const int hi = lane >> 4;

    // ---- A matrix (4-camera stack), uniform content in WMMA A layout
    const int acam = m >> 2, arow = m & 3;
    float a0 = 0.0f, a1 = 0.0f;
    if (arow < 3) {
        const float* ar = cams + acam * 16 + arow * 4 + kb;
        a0 = ar[0];
        a1 = ar[1];
    }
    v2f Am;
    Am.x = a0;
    Am.y = a1;

    // ---- cov3d for this lane's own gaussian (base+lane), no duplication
    const int gq = base + lane;
    float4 q = ((const float4*)quats)[gq];       // (r,x,y,z)
    float inv_nq = fast_rsqrt(q.x * q.x + q.y * q.y + q.z * q.z + q.w * q.w);
    float qr = q.x * inv_nq, qx = q.y * inv_nq, qy = q.z * inv_nq, qz = q.w * inv_nq;
    float R00 = 1.0f - 2.0f * (qy * qy + qz * qz);
    float R01 = 2.0f * (qx * qy - qr * qz);
    float R02 = 2.0f * (qx * qz + qr * qy);
    float R10 = 2.0f * (qx * qy + qr * qz);
    float R11 = 1.0f - 2.0f * (qx * qx + qz * qz);
    float R12 = 2.0f * (qy * qz - qr * qx);
    float R20 = 2.0f * (qx * qz - qr * qy);
    float R21 = 2.0f * (qy * qz + qr * qx);
    float R22 = 1.0f - 2.0f * (qx * qx + qy * qy);
    const float* sp = scales + 3 * (size_t)gq;
    float s0 = sp[0] * sp[0], s1 = sp[1] * sp[1], s2 = sp[2] * sp[2];
    float c00 = R00 * R00 * s0 + R01 * R01 * s1 + R02 * R02 * s2;
    float c01 = R00 * R10 * s0 + R01 * R11 * s1 + R02 * R12 * s2;
    float c02 = R00 * R20 * s0 + R01 * R21 * s1 + R02 * R22 * s2;
    float c11 = R10 * R10 * s0 + R11 * R11 * s1 + R12 * R12 * s2;
    float c12 = R10 * R20 * s0 + R11 * R21 * s1 + R12 * R22 * s2;
    float c22 = R20 * R20 * s0 + R21 * R21 * s1 + R22 * R22 * s2;

    // ---- camera constants for this lane's two cameras (hoisted to regs)
    const int cbase = hi << 1;           // 0 (lanes 0-15) or 2 (lanes 16-31)
    const float4* cwA = (const float4*)(cams + cbase * 16);
    float4 Ar0 = cwA[0], Ar1 = cwA[1], Ar2 = cwA[2], Afp = cwA[3];
    const float4* cwB = (const float4*)(cams + (cbase + 1) * 16);
    float4 Br0 = cwB[0], Br1 = cwB[1], Br2 = cwB[2], Bfp = cwB[3];

#pragma unroll
    for (int p = 0; p < 2; ++p) {
        const int tile = base + (p << 4);
        const int gp   = tile + m;       // gaussian this lane projects in phase p

        // ---- B matrix: homogeneous xyz columns of this 16-gaussian tile
        const float* gx = xyz + 3 * (size_t)gp;
        float b0, b1;
        if (lane < 16) { b0 = gx[0]; b1 = gx[1]; }   // K=0 (x), K=1 (y)
        else           { b0 = gx[2]; b1 = 1.0f; }    // K=2 (z), K=3 (ones)
        v2f Bm;
        Bm.x = b0;
        Bm.y = b1;

        v8f D = {};
        D = __builtin_amdgcn_wmma_f32_16x16x4_f32(
            false, Am, false, Bm, (short)0, D, false, false);

        // ---- pull cov3d of gaussian gp from owning lane (p*16 + m)
        const int src = (p << 4) + m;
        float t00 = lane_shuffle(src, c00);
        float t01 = lane_shuffle(src, c01);
        float t02 = lane_shuffle(src, c02);
        float t11 = lane_shuffle(src, c11);
        float t12 = lane_shuffle(src, c12);
        float t22 = lane_shuffle(src, c22);

#pragma unroll
        for (int s = 0; s < 2; ++s) {
            float mx = s ? D[4] : D[0];
            float my = s ? D[5] : D[1];
            float mz = s ? D[6] : D[2];
            float4 r0 = s ? Br0 : Ar0;
            float4 r1 = s ? Br1 : Ar1;
            float4 r2 = s ? Br2 : Ar2;
            float4 fp = s ? Bfp : Afp;
            float fx = fp.x, fy = fp.y, limx = fp.z, limy = fp.w;

            float inv_z = fast_rcp(mz);
            float tx = fminf(fmaxf(mx * inv_z, -limx), limx);
            float ty = fminf(fmaxf(my * inv_z, -limy), limy);
            float cx = tx * mz, cy = ty * mz;
            float iz2 = inv_z * inv_z;
            float j00 = fx * inv_z, j02 = -(fx * cx) * iz2;
            float j11 = fy * inv_z, j12 = -(fy * cy) * iz2;
            // T = J * Rcw (row 2 of J is zero)
            float T00 = j00 * r0.x + j02 * r2.x;
            float T01 = j00 * r0.y + j02 * r2.y;
            float T02 = j00 * r0.z + j02 * r2.z;
            float T10 = j11 * r1.x + j12 * r2.x;
            float T11 = j11 * r1.y + j12 * r2.y;
            float T12 = j11 * r1.z + j12 * r2.z;
            // M = T * C (C symmetric)
            float M00 = T00 * t00 + T01 * t01 + T02 * t02;
            float M01 = T00 * t01 + T01 * t11 + T02 * t12;
            float M02 = T00 * t02 + T01 * t12 + T02 * t22;
            float M10 = T10 * t00 + T11 * t01 + T12 * t02;
            float M11 = T10 * t01 + T11 * t11 + T12 * t12;
            float M12 = T10 * t02 + T11 * t12 + T12 * t22;
            // cov2d = M * T^T (2x2)
            float o00 = M00 * T00 + M01 * T01 + M02 * T02;
            float o01 = M00 * T10 + M01 * T11 + M02 * T12;
            float o10 = M10 * T00 + M11 * T01 + M12 * T02;
            float o11 = M10 * T10 + M11 * T11 + M12 * T12;

            ((float4*)out)[(size_t)(cbase + s) * n + gp] =
                make_float4(o00, o01, o10, o11);
        }
    }
}

extern "C" void kernel_launch(void* const* d_in, const int* in_sizes, int n_in,
                              void* d_out, int out_size, void* d_ws, size_t ws_size,
                              hipStream_t stream) {
    const int*   indexes = (const int*)d_in[0];
    const float* xyz     = (const float*)d_in[1];
    const float* scales  = (const float*)d_in[2];
    const float* quats   = (const float*)d_in[3];
    const float* tf      = (const float*)d_in[4];
    const float* focal   = (const float*)d_in[5];
    const float* widths  = (const float*)d_in[6];
    float* out  = (float*)d_out;
    float* cams = (float*)d_ws;   // 4 * 16 floats

    const int batch = in_sizes[0];
    const int n     = in_sizes[1] / 3;

    cam_setup_kernel<<<1, 64, 0, stream>>>(indexes, tf, focal, widths, cams, batch);

    const int waves   = (n + 31) / 32;               // 31250 for n = 1e6
    const long long threads = (long long)waves * 32;
    const int block   = 256;                         // 8 waves/block
    const int grid    = (int)((threads + block - 1) / block);
    splat_cov2d_kernel<<<grid, block, 0, stream>>>(xyz, scales, quats, cams, out, n);
}